// PeelerClass_65927747994106
// MI455X (gfx1250) — compile-verified
//
#include <hip/hip_runtime.h>

typedef __attribute__((ext_vector_type(2))) float v2f;
typedef __attribute__((ext_vector_type(8))) float v8f;

#define Bsz 512
#define Ksz 128
#define Csz 512
#define HW  49
#define HWP 52          // padded contraction length (13 * 4)
#define CCHUNK 64
#define NCCHUNK 8       // Csz / CCHUNK
#define NBTILE 32       // Bsz / 16

// One workgroup: 8 waves, one 16-row b-tile, one 64-channel c-chunk.
// Wave w computes the 16x16 output tile (btile, ktile=w) partial dist^2
// over its c-chunk. The X tile is double-buffered through LDS: staging of
// channel c+1 overlaps the 13 WMMAs of channel c, one barrier per channel.
__global__ __launch_bounds__(256) void pd_partial_kernel(
    const float* __restrict__ X, const float* __restrict__ sigs,
    const float* __restrict__ mu, float* __restrict__ ws)
{
    const int btile  = blockIdx.x & (NBTILE - 1);
    const int cchunk = blockIdx.x >> 5;
    const int tid  = threadIdx.x;
    const int lane = tid & 31;
    const int wv   = tid >> 5;          // k-tile id, 0..7
    const int hl   = lane >> 4;         // 0: lanes 0-15, 1: lanes 16-31
    const int l16  = lane & 15;

    __shared__ float ldsX[2][16 * HWP]; // 2 x 3328 B, rows of 52 (3 pad cols)

    const int b0 = btile * 16;
    const int kg = wv * 16 + l16;       // this lane's global k (B-matrix column)
    const float* sigrow = sigs + (size_t)kg * (Csz * HW);
    const float* murow  = mu   + (size_t)kg * Csz;

    // Static staging map: thread t -> row t>>4, base col t&15; each thread
    // loads hw = col, col+16, col+32 (and hw=48 for col==0). No divides.
    const int srow = tid >> 4;          // 0..15
    const int scol = tid & 15;          // 0..15
    const int cbase = cchunk * CCHUNK;
    const float* xrow0 = X + (size_t)(b0 + srow) * (Csz * HW) + (size_t)cbase * HW;
    float* lrow[2] = { &ldsX[0][srow * HWP], &ldsX[1][srow * HWP] };

    // Zero both buffers once (pad columns 49..51 stay zero forever).
    for (int idx = tid; idx < 2 * 16 * HWP; idx += 256)
        (&ldsX[0][0])[idx] = 0.f;
    __syncthreads();

    // Stage channel cbase into buffer 0.
    {
        float* lr = lrow[0];
        lr[scol]      = xrow0[scol];
        lr[scol + 16] = xrow0[scol + 16];
        lr[scol + 32] = xrow0[scol + 32];
        if (scol == 0) lr[48] = xrow0[48];
    }

    float dist2[8];
#pragma unroll
    for (int v = 0; v < 8; ++v) dist2[v] = 0.f;

    for (int cc = 0; cc < CCHUNK; ++cc) {
        __syncthreads();   // stage(cc) visible; compute(cc-1) finished

        // Prefetch channel cc+1 into the other buffer (overlaps WMMAs below).
        if (cc + 1 < CCHUNK) {
            const float* xr = xrow0 + (size_t)(cc + 1) * HW;
            float* lr = lrow[(cc + 1) & 1];
            lr[scol]      = xr[scol];
            lr[scol + 16] = xr[scol + 16];
            lr[scol + 32] = xr[scol + 32];
            if (scol == 0) lr[48] = xr[48];
        }

        const int c = cbase + cc;
        const float* sc   = sigrow + (size_t)c * HW;
        const float* arow = &ldsX[cc & 1][l16 * HWP];

        v8f acc = {};   // q-tile accumulator (16x16 f32 C/D layout)

        // hw = 0..47 : 12 full K=4 WMMAs
#pragma unroll
        for (int j = 0; j < 12; ++j) {
            const int hwA = 4 * j + hl * 2;
            v2f a = *(const v2f*)&arow[hwA];    // aligned ds_load_b64
            v2f b;
            b.x = sc[hwA];
            b.y = sc[hwA + 1];
            acc = __builtin_amdgcn_wmma_f32_16x16x4_f32(
                false, a, false, b, (short)0, acc, false, false);
        }
        // hw = 48 remainder (49..51 zero); cndmask instead of exec-divergence
        {
            v2f a = *(const v2f*)&arow[48 + hl * 2];
            const float t = sc[48];
            v2f b;
            b.x = hl ? 0.f : t;
            b.y = 0.f;
            acc = __builtin_amdgcn_wmma_f32_16x16x4_f32(
                false, a, false, b, (short)0, acc, false, false);
        }

        // Fold (q - mu)^2 for this channel into running dist^2.
        const float muv = murow[c];
#pragma unroll
        for (int v = 0; v < 8; ++v) {
            const float d = acc[v] * (1.0f / 49.0f) - muv;
            dist2[v] += d * d;
        }
    }

    // Write partial dist^2: ws[cchunk][b][k], each slot written exactly once.
    float* wsc = ws + (size_t)cchunk * (Bsz * Ksz);
#pragma unroll
    for (int v = 0; v < 8; ++v) {
        const int bg = b0 + v + hl * 8;    // C/D layout: M = v (+8 upper half)
        wsc[bg * Ksz + kg] = dist2[v];
    }
}

// Reduce the 8 c-chunk partials, out = -sqrt(dist^2).
__global__ __launch_bounds__(256) void pd_finish_kernel(
    const float* __restrict__ ws, float* __restrict__ out)
{
    const int i = blockIdx.x * 256 + threadIdx.x;
    float s = 0.f;
#pragma unroll
    for (int ch = 0; ch < NCCHUNK; ++ch)
        s += ws[(size_t)ch * (Bsz * Ksz) + i];
    out[i] = -__builtin_sqrtf(s);
}

extern "C" void kernel_launch(void* const* d_in, const int* in_sizes, int n_in,
                              void* d_out, int out_size, void* d_ws, size_t ws_size,
                              hipStream_t stream)
{
    const float* X    = (const float*)d_in[0];   // [512, 512, 7, 7]
    const float* sigs = (const float*)d_in[1];   // [128, 512, 7, 7]
    const float* mu   = (const float*)d_in[2];   // [128, 512]
    float* out = (float*)d_out;                  // [512, 128]
    float* ws  = (float*)d_ws;                   // 8 * 512 * 128 floats = 2 MB

    pd_partial_kernel<<<dim3(NBTILE * NCCHUNK), dim3(256), 0, stream>>>(X, sigs, mu, ws);
    pd_finish_kernel<<<dim3((Bsz * Ksz) / 256), dim3(256), 0, stream>>>(ws, out);
}